// CMmodel_54348516164068
// MI455X (gfx1250) — compile-verified
//
#include <hip/hip_runtime.h>
#include <hip/hip_bf16.h>

typedef __attribute__((ext_vector_type(2))) float v2f;
typedef __attribute__((ext_vector_type(8))) float v8f;

#define MEM_DIM   256
#define MEM_SIZE  2048
#define NROWS     32768
#define TOPK      10
#define BN_EPS    1e-5f
#define SLOPE     0.01f

#define XS_STRIDE  264     // 256 + 8 pad (even -> b64-aligned frags)
#define SIM_STRIDE 2052    // 2048 + 4 pad (8*2052 % 64 != 0 -> no half-wave bank clash)
#define NEG_INF   -3.4e38f

// ---------------- reciprocal row L2 norms: one wave per row ----------------
__global__ __launch_bounds__(256) void row_rnorm(const float* __restrict__ x,
                                                 float* __restrict__ rn) {
  int lane = threadIdx.x & 31, wave = threadIdx.x >> 5;
  int row  = blockIdx.x * 8 + wave;
  const float* xr = x + (size_t)row * MEM_DIM;
  float s = 0.f;
  #pragma unroll
  for (int d = lane; d < MEM_DIM; d += 32) { float v = xr[d]; s += v * v; }
  #pragma unroll
  for (int o = 16; o; o >>= 1) s += __shfl_xor(s, o, 32);
  if (lane == 0) rn[row] = rsqrtf(s);
}

// ---- fused: cosine-sim GEMM (WMMA f32) + top-k + softmax(relu) + gather ----
__global__ __launch_bounds__(256, 1) void fused_unit(const float* __restrict__ x,
                                                     const float* __restrict__ mem,
                                                     const float* __restrict__ rxn,
                                                     const float* __restrict__ rmn,
                                                     float* __restrict__ h) {
  extern __shared__ float smem[];
  float* Xs  = smem;                          // 16 x XS_STRIDE
  float* Sim = Xs + 16 * XS_STRIDE;           // 16 x SIM_STRIDE
  float* rx  = Sim + 16 * SIM_STRIDE;         // 16

  const int tid  = threadIdx.x;
  const int lane = tid & 31;
  const int wave = tid >> 5;
  const int row0 = blockIdx.x * 16;

  // cooperatively stage x tile (coalesced)
  for (int i = tid; i < 16 * MEM_DIM; i += 256) {
    int r = i >> 8, c = i & 255;
    Xs[r * XS_STRIDE + c] = x[(size_t)(row0 + r) * MEM_DIM + c];
  }
  if (tid < 16) rx[tid] = rxn[row0 + tid];
  __syncthreads();

  const int half = lane >> 4;     // K-pair selector per ISA A/B layout
  const int mn   = lane & 15;     // M for A-frag, N for B-frag

  // each wave owns 16 contiguous 16-wide column tiles
  const float* arow = Xs + mn * XS_STRIDE + 2 * half;
  for (int t = 0; t < 16; ++t) {
    const int j0 = (wave * 16 + t) * 16;
    const float* brow = mem + (size_t)(j0 + mn) * MEM_DIM + 2 * half;
    v8f acc = {0.f, 0.f, 0.f, 0.f, 0.f, 0.f, 0.f, 0.f};
    #pragma unroll 8
    for (int k0 = 0; k0 < MEM_DIM; k0 += 4) {
      v2f a = *(const v2f*)(arow + k0);   // ds_load_b64
      v2f b = *(const v2f*)(brow + k0);   // global_load_b64 (mem is L2-hot)
      acc = __builtin_amdgcn_wmma_f32_16x16x4_f32(false, a, false, b,
                                                  (short)0, acc, false, false);
    }
    // cosine scaling + spill tile into LDS sim buffer
    #pragma unroll
    for (int p = 0; p < 8; ++p) {
      int m = p + 8 * half;
      Sim[m * SIM_STRIDE + j0 + mn] = acc[p] * rx[m] * rmn[j0 + mn];
    }
  }
  __syncthreads();

  // each wave: top-10 + softmax(relu) + weighted gather for rows 2w, 2w+1
  for (int rr = 0; rr < 2; ++rr) {
    const int row = wave * 2 + rr;
    float* S = Sim + row * SIM_STRIDE;
    float topv[TOPK]; int topi[TOPK];
    for (int kk = 0; kk < TOPK; ++kk) {
      float best = NEG_INF; int bidx = 0x7fffffff;
      for (int j = lane; j < MEM_SIZE; j += 32) {
        float v = S[j];
        if (v > best) { best = v; bidx = j; }
      }
      #pragma unroll
      for (int o = 16; o; o >>= 1) {   // butterfly argmax -> all lanes agree
        float ov = __shfl_xor(best, o, 32);
        int   oi = __shfl_xor(bidx, o, 32);
        if (ov > best || (ov == best && oi < bidx)) { best = ov; bidx = oi; }
      }
      topv[kk] = best; topi[kk] = bidx;
      // punch out from the lane that will re-read it (same-wave LDS is in-order)
      if ((bidx & 31) == lane) S[bidx] = NEG_INF;
    }
    // softmax over relu'd scores (all lanes redundantly, registers only)
    float mx = 0.f;
    #pragma unroll
    for (int kk = 0; kk < TOPK; ++kk) {
      float v = topv[kk] > 0.f ? topv[kk] : 0.f;
      topv[kk] = v; mx = v > mx ? v : mx;
    }
    float sum = 0.f;
    #pragma unroll
    for (int kk = 0; kk < TOPK; ++kk) { float e = __expf(topv[kk] - mx); topv[kk] = e; sum += e; }
    const float inv = 1.f / sum;
    // weighted sum of gathered memory rows (coalesced over d)
    #pragma unroll
    for (int d = lane; d < MEM_DIM; d += 32) {
      float o = 0.f;
      #pragma unroll
      for (int kk = 0; kk < TOPK; ++kk)
        o += topv[kk] * mem[(size_t)topi[kk] * MEM_DIM + d];
      h[(size_t)(row0 + row) * MEM_DIM + d] = o * inv;
    }
  }
}

// ------------- deterministic BN stats: fixed-tree column reduction ----------
__global__ __launch_bounds__(256) void colstats_partial(const float* __restrict__ h,
                                                        float* __restrict__ psum,
                                                        float* __restrict__ psq) {
  const int col = threadIdx.x;            // 256 columns
  const int r0  = blockIdx.x * 256;       // 128 chunks of 256 rows
  float s = 0.f, q = 0.f;
  for (int r = 0; r < 256; ++r) {
    float v = h[(size_t)(r0 + r) * MEM_DIM + col];
    s += v; q += v * v;
  }
  psum[blockIdx.x * MEM_DIM + col] = s;
  psq [blockIdx.x * MEM_DIM + col] = q;
}

__global__ __launch_bounds__(256) void colstats_final(const float* __restrict__ psum,
                                                      const float* __restrict__ psq,
                                                      float* __restrict__ mean,
                                                      float* __restrict__ rstd) {
  const int col = threadIdx.x;
  float s = 0.f, q = 0.f;
  for (int c = 0; c < 128; ++c) { s += psum[c * MEM_DIM + col]; q += psq[c * MEM_DIM + col]; }
  const float m   = s * (1.f / NROWS);
  const float var = q * (1.f / NROWS) - m * m;
  mean[col] = m;
  rstd[col] = rsqrtf(var + BN_EPS);
}

__global__ __launch_bounds__(256) void bn_leaky(float* __restrict__ h,
                                                const float* __restrict__ gamma,
                                                const float* __restrict__ beta,
                                                const float* __restrict__ mean,
                                                const float* __restrict__ rstd) {
  const int idx = blockIdx.x * 256 + threadIdx.x;
  const int col = idx & 255;
  float v = h[idx];
  v = gamma[col] * (v - mean[col]) * rstd[col] + beta[col];
  h[idx] = v >= 0.f ? v : SLOPE * v;
}

extern "C" void kernel_launch(void* const* d_in, const int* in_sizes, int n_in,
                              void* d_out, int out_size, void* d_ws, size_t ws_size,
                              hipStream_t stream) {
  const float* x      = (const float*)d_in[0];
  const float* mem1   = (const float*)d_in[1];
  const float* mem2   = (const float*)d_in[2];
  const float* gamma1 = (const float*)d_in[3];
  const float* beta1  = (const float*)d_in[4];
  const float* gamma2 = (const float*)d_in[5];
  const float* beta2  = (const float*)d_in[6];
  float* out = (float*)d_out;

  // workspace carve (~34 MB)
  float* h1   = (float*)d_ws;                       // 32768*256
  float* rxn  = h1   + (size_t)NROWS * MEM_DIM;     // 32768
  float* rhn  = rxn  + NROWS;                       // 32768
  float* rmn1 = rhn  + NROWS;                       // 2048
  float* rmn2 = rmn1 + MEM_SIZE;                    // 2048
  float* psum = rmn2 + MEM_SIZE;                    // 128*256
  float* psq  = psum + 128 * MEM_DIM;               // 128*256
  float* mean = psq  + 128 * MEM_DIM;               // 256
  float* rstd = mean + MEM_DIM;                     // 256

  const size_t smbytes = (size_t)(16 * XS_STRIDE + 16 * SIM_STRIDE + 16) * sizeof(float);
  (void)hipFuncSetAttribute((const void*)fused_unit,
                            hipFuncAttributeMaxDynamicSharedMemorySize, (int)smbytes);

  // layer 1
  row_rnorm<<<NROWS / 8, 256, 0, stream>>>(x, rxn);
  row_rnorm<<<MEM_SIZE / 8, 256, 0, stream>>>(mem1, rmn1);
  row_rnorm<<<MEM_SIZE / 8, 256, 0, stream>>>(mem2, rmn2);
  fused_unit<<<NROWS / 16, 256, smbytes, stream>>>(x, mem1, rxn, rmn1, h1);
  colstats_partial<<<128, 256, 0, stream>>>(h1, psum, psq);
  colstats_final<<<1, 256, 0, stream>>>(psum, psq, mean, rstd);
  bn_leaky<<<(NROWS * MEM_DIM) / 256, 256, 0, stream>>>(h1, gamma1, beta1, mean, rstd);

  // layer 2
  row_rnorm<<<NROWS / 8, 256, 0, stream>>>(h1, rhn);
  fused_unit<<<NROWS / 16, 256, smbytes, stream>>>(h1, mem2, rhn, rmn2, out);
  colstats_partial<<<128, 256, 0, stream>>>(out, psum, psq);
  colstats_final<<<1, 256, 0, stream>>>(psum, psq, mean, rstd);
  bn_leaky<<<(NROWS * MEM_DIM) / 256, 256, 0, stream>>>(out, gamma2, beta2, mean, rstd);
}